// RecurrentEvolveGCNO_80814104641669
// MI455X (gfx1250) — compile-verified
//
#include <hip/hip_runtime.h>
#include <math.h>

typedef float v2f __attribute__((ext_vector_type(2)));
typedef float v8f __attribute__((ext_vector_type(8)));

#define NN 50000
#define NE 800000
#define FD 128

// ---------------------------------------------------------------- zero init
__global__ void zero_kernel(float* __restrict__ p, int n) {
    int i = blockIdx.x * blockDim.x + threadIdx.x;
    if (i < n) p[i] = 0.0f;
}

// ---------------------------------------------------------------- GRU evolve
// W[i][j]: gi = W0 @ W_ih^T + b_ih ; gh = W0 @ W_hh^T + b_hh
// r=sig(i_r+h_r); z=sig(i_z+h_z); n=tanh(i_n + r*h_n); W=(1-z)*n + z*W0
__global__ void evolve_kernel(const float* __restrict__ W0,
                              const float* __restrict__ W_ih,
                              const float* __restrict__ W_hh,
                              const float* __restrict__ b_ih,
                              const float* __restrict__ b_hh,
                              float* __restrict__ Wout) {
    __shared__ float w0row[FD];
    const int i = blockIdx.x;   // output row
    const int j = threadIdx.x;  // output col
    w0row[j] = W0[i * FD + j];
    __syncthreads();

    float g[6];
#pragma unroll
    for (int t = 0; t < 6; ++t) {
        const float* Wt = (t < 3) ? W_ih : W_hh;
        const int gate = t % 3;
        const float* row = Wt + (size_t)(gate * FD + j) * FD;
        float acc = 0.0f;
        for (int k = 0; k < FD; ++k) acc += w0row[k] * row[k];
        acc += ((t < 3) ? b_ih : b_hh)[gate * FD + j];
        g[t] = acc;
    }
    const float r = 1.0f / (1.0f + expf(-(g[0] + g[3])));
    const float z = 1.0f / (1.0f + expf(-(g[1] + g[4])));
    const float n = tanhf(g[2] + r * g[5]);
    Wout[i * FD + j] = (1.0f - z) * n + z * w0row[j];
}

// ---------------------------------------------------------------- xw = x @ W
// Block = 256 threads = 8 wave32s; block computes a 16x128 strip of xw.
// Full W (128x128) staged in LDS (stride-132 padded), x strip 16x128 in LDS.
// Each wave: one 16x16 C tile, K-loop of 32 x v_wmma_f32_16x16x4_f32.
#define SBS 132  // padded LDS row stride (floats)
__global__ void __launch_bounds__(256)
gemm_xw_kernel(const float* __restrict__ x, const float* __restrict__ W,
               float* __restrict__ xw) {
    extern __shared__ float smem[];
    float* sB = smem;              // 128 x SBS  (W, row-major K x N)
    float* sA = smem + FD * SBS;   // 16 x SBS   (x strip, row-major M x K)

    const int tid = threadIdx.x;
    const int m0  = blockIdx.x * 16;

    for (int idx = tid; idx < FD * FD; idx += 256) {
        int r = idx >> 7, c = idx & 127;
        sB[r * SBS + c] = W[idx];
    }
    for (int idx = tid; idx < 16 * FD; idx += 256) {
        int r = idx >> 7, c = idx & 127;
        sA[r * SBS + c] = x[(size_t)(m0 + r) * FD + c];
    }
    __syncthreads();

    const int lane  = tid & 31;
    const int wave  = tid >> 5;
    const int n0    = wave * 16;          // this wave's column slice of W
    const int lrow  = lane & 15;
    const int khalf = (lane < 16) ? 0 : 2;  // ISA A/B fragment lane split

    v8f c = {0.f, 0.f, 0.f, 0.f, 0.f, 0.f, 0.f, 0.f};
#pragma unroll 4
    for (int k0 = 0; k0 < FD; k0 += 4) {
        const int kb = k0 + khalf;
        v2f a, b;
        // A 16x4 (MxK): lane L holds row M=L%16; VGPR0/1 = K {kb, kb+1}
        a.x = sA[lrow * SBS + kb];
        a.y = sA[lrow * SBS + kb + 1];
        // B 4x16 (KxN): lane L holds col N=L%16; VGPR0/1 = K {kb, kb+1}
        b.x = sB[kb * SBS + n0 + lrow];
        b.y = sB[(kb + 1) * SBS + n0 + lrow];
        c = __builtin_amdgcn_wmma_f32_16x16x4_f32(
                /*neg_a=*/false, a, /*neg_b=*/false, b,
                /*c_mod=*/(short)0, c, /*reuse_a=*/false, /*reuse_b=*/false);
    }
    // C 16x16 f32: VGPR v -> row v (lanes 0-15) or v+8 (lanes 16-31), col lane%16
    const int crow = (lane < 16) ? 0 : 8;
#pragma unroll
    for (int v = 0; v < 8; ++v)
        xw[(size_t)(m0 + crow + v) * FD + n0 + lrow] = c[v];
}

// ---------------------------------------------------------------- degree
__global__ void deg_kernel(const int* __restrict__ dst,
                           const float* __restrict__ ew,
                           float* __restrict__ deg) {
    int i = blockIdx.x * blockDim.x + threadIdx.x;
    if (i < NE)            atomicAdd(&deg[dst[i]], ew[i]);
    else if (i < NE + NN)  atomicAdd(&deg[i - NE], 1.0f);  // self loops, w=1
}

__global__ void dinv_kernel(float* __restrict__ deg_dinv) {
    int i = blockIdx.x * blockDim.x + threadIdx.x;
    if (i < NN) {
        float d = deg_dinv[i];
        deg_dinv[i] = (d > 0.0f) ? rsqrtf(d) : 0.0f;
    }
}

// ---------------------------------------------------------------- edge scatter
// One wave per edge; lane l carries features [4l, 4l+4). xw/h live in L2.
__global__ void scatter_kernel(const int* __restrict__ src,
                               const int* __restrict__ dst,
                               const float* __restrict__ ew,
                               const float* __restrict__ dinv,
                               const float* __restrict__ xw,
                               float* __restrict__ h) {
    const int gid  = blockIdx.x * blockDim.x + threadIdx.x;
    const int wid  = gid >> 5;
    const int lane = gid & 31;
    const int nwav = (gridDim.x * blockDim.x) >> 5;
    for (int e = wid; e < NE; e += nwav) {
        const int s = src[e], d = dst[e];
        const float norm = dinv[s] * ew[e] * dinv[d];
        const float4 v = reinterpret_cast<const float4*>(xw + (size_t)s * FD)[lane];
        float* hr = h + (size_t)d * FD + lane * 4;
        atomicAdd(hr + 0, norm * v.x);
        atomicAdd(hr + 1, norm * v.y);
        atomicAdd(hr + 2, norm * v.z);
        atomicAdd(hr + 3, norm * v.w);
    }
}

// ---------------------------------------------------------------- finalize
// h += dinv[n]^2 * xw[n] (self loop), tanh, dot with w_lin, +b_lin.
__global__ void finalize_kernel(const float* __restrict__ h,
                                const float* __restrict__ xw,
                                const float* __restrict__ dinv,
                                const float* __restrict__ w_lin,
                                const float* __restrict__ b_lin,
                                float* __restrict__ out) {
    const int gid  = blockIdx.x * blockDim.x + threadIdx.x;
    const int node = gid >> 5;
    const int lane = gid & 31;
    if (node >= NN) return;
    const float di = dinv[node];
    const float self = di * di;
    const float4 hv = reinterpret_cast<const float4*>(h  + (size_t)node * FD)[lane];
    const float4 xv = reinterpret_cast<const float4*>(xw + (size_t)node * FD)[lane];
    const float4 wv = reinterpret_cast<const float4*>(w_lin)[lane];
    float acc = tanhf(hv.x + self * xv.x) * wv.x
              + tanhf(hv.y + self * xv.y) * wv.y
              + tanhf(hv.z + self * xv.z) * wv.z
              + tanhf(hv.w + self * xv.w) * wv.w;
#pragma unroll
    for (int off = 16; off > 0; off >>= 1)
        acc += __shfl_down(acc, off, 32);
    if (lane == 0) out[node] = acc + b_lin[0];
}

// ---------------------------------------------------------------- launch
extern "C" void kernel_launch(void* const* d_in, const int* in_sizes, int n_in,
                              void* d_out, int out_size, void* d_ws, size_t ws_size,
                              hipStream_t stream) {
    const float* x    = (const float*)d_in[0];
    const int*   ei   = (const int*)  d_in[1];   // (2,E): [0,E)=src, [E,2E)=dst
    const float* ew   = (const float*)d_in[2];
    const float* W0   = (const float*)d_in[3];
    const float* Wih  = (const float*)d_in[4];
    const float* Whh  = (const float*)d_in[5];
    const float* bih  = (const float*)d_in[6];
    const float* bhh  = (const float*)d_in[7];
    const float* wlin = (const float*)d_in[8];
    const float* blin = (const float*)d_in[9];
    float* out = (float*)d_out;
    float* ws  = (float*)d_ws;

    const int* src = ei;
    const int* dst = ei + NE;

    // workspace layout (floats, 16B-aligned chunks)
    float* W    = ws;                         // 16384
    float* dinv = ws + 16384;                 // 50000 (deg, then dinv in place)
    float* xw   = ws + 16384 + 50016;         // 6.4M
    float* h    = xw + (size_t)NN * FD;       // 6.4M

    // re-zero atomic accumulators every call (graph replay determinism)
    zero_kernel<<<(NN + 255) / 256, 256, 0, stream>>>(dinv, NN);
    zero_kernel<<<(NN * FD + 255) / 256, 256, 0, stream>>>(h, NN * FD);

    evolve_kernel<<<FD, FD, 0, stream>>>(W0, Wih, Whh, bih, bhh, W);

    deg_kernel<<<(NE + NN + 255) / 256, 256, 0, stream>>>(dst, ew, dinv);
    dinv_kernel<<<(NN + 255) / 256, 256, 0, stream>>>(dinv);

    const size_t shmem = (size_t)(FD * SBS + 16 * SBS) * sizeof(float); // ~76 KB < 320 KB/WGP
    gemm_xw_kernel<<<NN / 16, 256, shmem, stream>>>(x, W, xw);

    const int scatterBlocks = (NE * 32 + 255) / 256;  // one wave per edge
    scatter_kernel<<<scatterBlocks, 256, 0, stream>>>(src, dst, ew, dinv, xw, h);

    finalize_kernel<<<(NN * 32 + 255) / 256, 256, 0, stream>>>(h, xw, dinv, wlin, blin, out);
}